// GCNGRUCell_31791347925777
// MI455X (gfx1250) — compile-verified
//
#include <hip/hip_runtime.h>
#include <hip/hip_bf16.h>
#include <math.h>

// ---------------------------------------------------------------------------
// GCN-GRU cell for MI455X (gfx1250), fp32 via V_WMMA_F32_16X16X4_F32.
// NT GEMM: both operands K-contiguous; LDS staged with CDNA5 async
// global->LDS b128 copies (ASYNCcnt), double-buffered.
// B=64, N=2048, H=64, C=B*65=4160 columns in the propagation GEMMs.
// ---------------------------------------------------------------------------

typedef float v2f __attribute__((ext_vector_type(2)));
typedef float v8f __attribute__((ext_vector_type(8)));

#define NN   2048      // nodes (= K of the propagation GEMMs)
#define BB   64        // batch
#define HH   64        // hidden
#define CC   4160      // B*(H+1) columns of X / D
#define TM   128       // GEMM tile M
#define TN   64        // GEMM tile N
#define TK   32        // GEMM K stage
#define NSTG (NN / TK) // 64 K-stages

// ---- CDNA5 async global->LDS (ASYNCcnt) helpers ---------------------------
__device__ __forceinline__ void async_copy_b128(void* lds_ptr, const float* gptr) {
    unsigned lds_off = (unsigned)(unsigned long long)lds_ptr;  // addr[31:0] = LDS offset
    unsigned long long ga = (unsigned long long)gptr;
    asm volatile("global_load_async_to_lds_b128 %0, %1, off"
                 :: "v"(lds_off), "v"(ga) : "memory");
}
__device__ __forceinline__ void wait_async_zero() {
#if __has_builtin(__builtin_amdgcn_s_wait_asynccnt)
    __builtin_amdgcn_s_wait_asynccnt(0);
#else
    asm volatile("s_wait_asynccnt 0x0" ::: "memory");
#endif
}

// --------------------------- d = (rowsum+1)^-0.5 ---------------------------
__global__ __launch_bounds__(256)
void rowsum_rsqrt_kernel(const float* __restrict__ adj, float* __restrict__ dv) {
    __shared__ float red[256];
    const int row = blockIdx.x;
    float s = 0.f;
    for (int j = threadIdx.x; j < NN; j += 256) s += adj[(size_t)row * NN + j];
    red[threadIdx.x] = s;
    __syncthreads();
    for (int off = 128; off > 0; off >>= 1) {
        if (threadIdx.x < off) red[threadIdx.x] += red[threadIdx.x + off];
        __syncthreads();
    }
    if (threadIdx.x == 0) {
        float t = red[0] + 1.0f;  // +I diagonal
        dv[row] = (t > 0.f) ? rsqrtf(t) : 0.f;
    }
}

// ---------------- Anorm[m,n] = d[m] * (adj[n,m] + (m==n)) * d[n] -----------
__global__ __launch_bounds__(256)
void build_A_kernel(const float* __restrict__ adj, const float* __restrict__ dv,
                    float* __restrict__ An) {
    const int idx = blockIdx.x * 256 + threadIdx.x;
    const int m = idx >> 11;
    const int n = idx & (NN - 1);
    float a = adj[(size_t)n * NN + m] + ((m == n) ? 1.0f : 0.0f);
    An[idx] = dv[m] * dv[n] * a;
}

// ---- pack K-transposed activations: Xt[(b*65+c)*NN + n] -------------------
// c==0 -> input[b,n]; c>=1 -> T[b,n,c-1]. LDS tile transpose for coalescing.
// grid: (N/64, B); block: 256.
__global__ __launch_bounds__(256)
void pack_xt_kernel(const float* __restrict__ input, const float* __restrict__ T,
                    float* __restrict__ Xt) {
    __shared__ float buf[64][66];
    const int n0 = blockIdx.x * 64;
    const int b  = blockIdx.y;

    // phase 1: load 64 nodes x (1 + 64) channels, coalesced in h
    #pragma unroll
    for (int i = 0; i < 16; i++) {
        const int idx = i * 256 + threadIdx.x;      // 0..4095
        const int nl = idx >> 6, h = idx & 63;
        buf[nl][h + 1] = T[((size_t)b * NN + n0 + nl) * HH + h];
    }
    if (threadIdx.x < 64)
        buf[threadIdx.x][0] = input[(size_t)b * NN + n0 + threadIdx.x];
    __syncthreads();

    // phase 2: write 65 columns x 64 nodes, coalesced in n
    for (int i = 0; i < 17; i++) {
        const int idx = i * 256 + threadIdx.x;      // 0..4351
        if (idx < 65 * 64) {
            const int c = idx >> 6, nl = idx & 63;
            Xt[((size_t)b * 65 + c) * NN + n0 + nl] = buf[nl][c];
        }
    }
}

// ------------------- D = An (MxK) @ Xt^T  (Xt is CxK) ----------------------
// 128 threads = 4 waves (2 M x 2 N), each wave computes 64x32 output as a
// 4x2 grid of 16x16 f32 WMMA accumulators. Both LDS tiles are K-contiguous,
// so A and B fragments are identical contiguous b64 loads.
__global__ __launch_bounds__(128)
void gemm_f32_wmma_kernel(const float* __restrict__ A, const float* __restrict__ Xt,
                          float* __restrict__ D) {
    __shared__ float As[2][TM][TK + 4];   // row stride 36 floats (144B, 16B-aligned)
    __shared__ float Bs[2][TN][TK + 4];

    const int tid  = threadIdx.x;
    const int m0   = blockIdx.x * TM;
    const int n0   = blockIdx.y * TN;
    const int wave = tid >> 5;
    const int lane = tid & 31;
    const int half = lane >> 4;    // lane half (0/1)
    const int lm   = lane & 15;
    const int wmB  = (wave >> 1) * 64;   // wave M base within tile
    const int wnB  = (wave & 1) * 32;    // wave N base within tile

    v8f acc[4][2];
    #pragma unroll
    for (int i = 0; i < 4; i++)
        #pragma unroll
        for (int j = 0; j < 2; j++)
            #pragma unroll
            for (int v = 0; v < 8; v++) acc[i][j][v] = 0.0f;

    // ---- async-stage one K-slice into buffer `buf` ----
    auto stage = [&](int s, int buf) {
        const int k0 = s * TK;
        // A tile: 128x32 floats = 1024 b128, 8 per thread, lane-major (coalesced)
        #pragma unroll
        for (int i = 0; i < 8; i++) {
            const int q  = i * 128 + tid;       // 0..1023
            const int r  = q >> 3;              // row 0..127
            const int c4 = (q & 7) * 4;         // col 0..28
            async_copy_b128(&As[buf][r][c4], A + (size_t)(m0 + r) * NN + k0 + c4);
        }
        // B tile: 64 rows (cols of output) x 32 k = 512 b128, 4 per thread
        #pragma unroll
        for (int i = 0; i < 4; i++) {
            const int q  = i * 128 + tid;       // 0..511
            const int r  = q >> 3;              // row 0..63
            const int c4 = (q & 7) * 4;         // col 0..28
            async_copy_b128(&Bs[buf][r][c4], Xt + (size_t)(n0 + r) * NN + k0 + c4);
        }
    };

    stage(0, 0);

    for (int s = 0; s < NSTG; ++s) {
        const int buf = s & 1;
        wait_async_zero();      // own stage-s copies complete (in-order)
        __syncthreads();        // everyone's copies complete; prev buf free
        if (s + 1 < NSTG) stage(s + 1, buf ^ 1);

        #pragma unroll
        for (int kk = 0; kk < TK; kk += 4) {
            v2f af[4];
            v2f bf[2];
            #pragma unroll
            for (int tm = 0; tm < 4; tm++)
                af[tm] = *(const v2f*)&As[buf][wmB + tm * 16 + lm][kk + 2 * half];
            #pragma unroll
            for (int tn = 0; tn < 2; tn++)
                bf[tn] = *(const v2f*)&Bs[buf][wnB + tn * 16 + lm][kk + 2 * half];
            #pragma unroll
            for (int tm = 0; tm < 4; tm++)
                #pragma unroll
                for (int tn = 0; tn < 2; tn++)
                    acc[tm][tn] = __builtin_amdgcn_wmma_f32_16x16x4_f32(
                        false, af[tm], false, bf[tn], (short)0, acc[tm][tn],
                        false, false);
        }
        __syncthreads();        // done reading buf before it is restaged
    }

    // ---- store: C/D layout: VGPR v, lanes 0-15 -> M=v, lanes 16-31 -> M=8+v
    #pragma unroll
    for (int tm = 0; tm < 4; tm++)
        #pragma unroll
        for (int tn = 0; tn < 2; tn++) {
            const int mb = m0 + wmB + tm * 16 + half * 8;
            const int nb = n0 + wnB + tn * 16 + lm;
            #pragma unroll
            for (int v = 0; v < 8; v++)
                D[(size_t)(mb + v) * CC + nb] = acc[tm][tn][v];
        }
}

// ----- epilogue 1: sigmoid(D1 @ W1 + b1), node-major chunk -> RH / U -------
__global__ __launch_bounds__(256)
void epi1_kernel(const float* __restrict__ D1, const float* __restrict__ hidden,
                 const float* __restrict__ W1, const float* __restrict__ b1,
                 float* __restrict__ RH, float* __restrict__ U) {
    __shared__ float w[65 * 128];
    __shared__ float bb[128];
    __shared__ float rows[2][65];
    for (int i = threadIdx.x; i < 65 * 128; i += 256) w[i] = W1[i];
    if (threadIdx.x < 128) bb[threadIdx.x] = b1[threadIdx.x];
    __syncthreads();

    const int sub = threadIdx.x >> 7;      // which of 2 rows
    const int c1  = threadIdx.x & 127;
    const int base = blockIdx.x * 64;      // 64 (b,m) rows per block

    for (int r = 0; r < 64; r += 2) {
        if (threadIdx.x < 130) {
            const int rr = threadIdx.x / 65, cc = threadIdx.x % 65;
            const int rowL = base + r + rr;
            const int bL = rowL >> 11, mL = rowL & (NN - 1);
            rows[rr][cc] = D1[(size_t)mL * CC + bL * 65 + cc];
        }
        __syncthreads();

        float acc = bb[c1];
        #pragma unroll 5
        for (int c = 0; c < 65; c++) acc = fmaf(rows[sub][c], w[c * 128 + c1], acc);
        const float sig = 1.0f / (1.0f + __expf(-acc));

        const int row = base + r + sub;
        const int b = row >> 11, m = row & (NN - 1);
        if (m < NN / 2) {
            // r-gate: hidden node n = 2m + (c1>=64), channel h = c1 & 63
            const int n = 2 * m + (c1 >> 6), h = c1 & 63;
            const size_t hi = ((size_t)b * NN + n) * HH + h;
            RH[hi] = sig * hidden[hi];
        } else {
            // u-gate: flat index k into (N*H) node-major vector
            const int k = (m - NN / 2) * 128 + c1;
            U[(size_t)b * (NN * HH) + k] = sig;
        }
        __syncthreads();
    }
}

// ----- epilogue 2: c = tanh(D2 @ W2 + b2); h' = u*h + (1-u)*c --------------
__global__ __launch_bounds__(256)
void epi2_kernel(const float* __restrict__ D2, const float* __restrict__ hidden,
                 const float* __restrict__ W2, const float* __restrict__ b2,
                 const float* __restrict__ U, float* __restrict__ out) {
    __shared__ float w[65 * 64];
    __shared__ float bb[64];
    __shared__ float rows[4][65];
    for (int i = threadIdx.x; i < 65 * 64; i += 256) w[i] = W2[i];
    if (threadIdx.x < 64) bb[threadIdx.x] = b2[threadIdx.x];
    __syncthreads();

    const int sub = threadIdx.x >> 6;     // which of 4 rows
    const int h   = threadIdx.x & 63;
    const int base = blockIdx.x * 64;

    for (int r = 0; r < 64; r += 4) {
        for (int i = threadIdx.x; i < 4 * 65; i += 256) {
            const int rr = i / 65, cc = i % 65;
            const int rowL = base + r + rr;
            const int bL = rowL >> 11, mL = rowL & (NN - 1);
            rows[rr][cc] = D2[(size_t)mL * CC + bL * 65 + cc];
        }
        __syncthreads();

        const int row = base + r + sub;
        float acc = bb[h];
        #pragma unroll 5
        for (int c = 0; c < 65; c++) acc = fmaf(rows[sub][c], w[c * 64 + h], acc);
        const float ct = tanhf(acc);

        const size_t hi = (size_t)row * HH + h;   // (b*N+m)*H + h
        const float u  = U[hi];
        const float hv = hidden[hi];
        out[hi] = u * hv + (1.0f - u) * ct;
        __syncthreads();
    }
}

// ---------------------------------------------------------------------------
extern "C" void kernel_launch(void* const* d_in, const int* in_sizes, int n_in,
                              void* d_out, int out_size, void* d_ws, size_t ws_size,
                              hipStream_t stream) {
    const float* input  = (const float*)d_in[0];  // (B,N)
    const float* hidden = (const float*)d_in[1];  // (B,N,H)
    const float* adj    = (const float*)d_in[2];  // (N,N)
    const float* W1     = (const float*)d_in[3];  // (65,128)
    const float* b1     = (const float*)d_in[4];  // (128,)
    const float* W2     = (const float*)d_in[5];  // (65,64)
    const float* b2     = (const float*)d_in[6];  // (64,)
    float* out = (float*)d_out;

    float* ws = (float*)d_ws;
    float* An = ws;                                   // N*N   = 4,194,304
    float* Xt = An + (size_t)NN * NN;                 // CC*N  = 8,519,680
    float* D  = Xt + (size_t)CC * NN;                 // N*CC  = 8,519,680
    float* U  = D  + (size_t)NN * CC;                 // B*N*H = 8,388,608
    float* RH = U  + (size_t)BB * NN * HH;            // B*N*H = 8,388,608
    float* dv = RH + (size_t)BB * NN * HH;            // N     = 2,048

    // 1) degree^{-1/2}
    rowsum_rsqrt_kernel<<<NN, 256, 0, stream>>>(adj, dv);
    // 2) normalized adjacency
    build_A_kernel<<<(NN * NN) / 256, 256, 0, stream>>>(adj, dv, An);
    // 3) pack conc1 (K-transposed)
    pack_xt_kernel<<<dim3(NN / 64, BB), 256, 0, stream>>>(input, hidden, Xt);
    // 4) D1 = An @ conc1   (WMMA f32, async-LDS staged, NT)
    gemm_f32_wmma_kernel<<<dim3(NN / TM, CC / TN), 128, 0, stream>>>(An, Xt, D);
    // 5) gates: RH = r*h (node-major), U = u
    epi1_kernel<<<(BB * NN) / 64, 256, 0, stream>>>(D, hidden, W1, b1, RH, U);
    // 6) pack conc2 from RH, then D2 = An @ conc2
    pack_xt_kernel<<<dim3(NN / 64, BB), 256, 0, stream>>>(input, RH, Xt);
    gemm_f32_wmma_kernel<<<dim3(NN / TM, CC / TN), 128, 0, stream>>>(An, Xt, D);
    // 7) candidate + GRU combine
    epi2_kernel<<<(BB * NN) / 64, 256, 0, stream>>>(D, hidden, W2, b2, U, out);
}